// BaconMultiSumHead_52192442581044
// MI455X (gfx1250) — compile-verified
//
#include <hip/hip_runtime.h>
#include <stdint.h>

// ---------------------------------------------------------------------------
// Soft decision-tree forest forward (BaconMultiSumHead), MI455X / gfx1250.
// VALU-bound (~100x over the 156MB @ 23.3TB/s memory time), so the compute
// is reduced to 3 VALU/node via the identity
//     w*min(a,b)+(1-w)*max(a,b) = (a+b)/2 + (0.5-w)*|a-b|
// with exact power-of-two layer scaling:  y' = (a+b) + c*|a-b|, c = 1-2*sigm.
// Layer-1 sums/diffs are shared across all 19 trees. Weights go through the
// scalar (SMEM) path; data is staged with the CDNA5 async global<->LDS mover.
// ---------------------------------------------------------------------------

#define BLK     320          // 320 | 1,000,000 -> no tail for the main kernel
#define IN_W    20
#define OUT_W   19
#define NTREE   19
#define CSTRIDE 32           // padded coeff row stride (128B-aligned rows)

// ---- Kernel 1: c[t][k] = 1 - 2*sigmoid(w_raw[t*19+k]), padded layout ------
__global__ void sigmoid_prep_kernel(const float* __restrict__ w_raw,
                                    float* __restrict__ cw)
{
    int j = blockIdx.x * blockDim.x + threadIdx.x;
    if (j < NTREE * 19) {
        int t = j / 19, k = j - t * 19;
        float s = 1.0f / (1.0f + expf(-w_raw[j]));
        cw[t * CSTRIDE + k] = 1.0f - 2.0f * s;     // c = 1 - 2w  in (-1,1)
    }
}

// One scaled node:  y' = (a+b) + c*|a-b|   (2x the true node value)
__device__ __forceinline__ float node2x(float a, float b, float c)
{
    return fmaf(c, fabsf(a - b), a + b);   // v_sub + v_fma(|.|) ; add inside fma src
}

// Evaluate all 19 trees for one sample. s1/d1 are the shared layer-1
// sums/diffs (scale 1 inputs). dst[t] receives the clipped outputs.
__device__ __forceinline__ void eval_forest(const float s1[10], const float d1[10],
                                            const float* __restrict__ cw,
                                            float* dst)
{
    #pragma unroll
    for (int t = 0; t < NTREE; ++t) {
        const float* __restrict__ ct = cw + t * CSTRIDE;
        // layer 1 (shared adds/subs): y1 = s1 + c*|d1|          scale 2
        float y1[10];
        #pragma unroll
        for (int j = 0; j < 10; ++j)
            y1[j] = fmaf(ct[j], fabsf(d1[j]), s1[j]);
        // layer 2: 10 -> 5                                      scale 4
        float y2[5];
        #pragma unroll
        for (int j = 0; j < 5; ++j)
            y2[j] = node2x(y1[2*j], y1[2*j+1], ct[10 + j]);
        // layer 3: 5 -> 2 (+carry y2[4])                        scale 8
        float y3a = node2x(y2[0], y2[1], ct[15]);
        float y3b = node2x(y2[2], y2[3], ct[16]);
        // layer 4                                               scale 16
        float y4  = node2x(y3a, y3b, ct[17]);
        // carry y2[4] is scale 4 -> align to 16 (exact)
        float c4  = y2[4] * 4.0f;
        // layer 5                                               scale 32
        float y5  = node2x(y4, c4, ct[18]);
        float r   = y5 * 0.03125f;                               // exact /32
        r = fminf(fmaxf(r, 1e-6f), 1.0f - 1e-6f);
        dst[t] = r;
    }
}

// ---- Kernel 2: main, 320 samples/block, async LDS staging -----------------
__global__ __launch_bounds__(BLK) void tree_forward_kernel(
        const float* __restrict__ c20,
        const float* __restrict__ cw_in,
        float* __restrict__ out)
{
    const float* __restrict__ cw =
        (const float*)__builtin_assume_aligned(cw_in, 256);

    // 320*20 = 6400 dwords in; reused for 320*19 = 6080 dwords out.
    __shared__ __align__(16) float tile[BLK * IN_W];
    const int tid = threadIdx.x;

    // -- Stage 1: async copy 320x20 f32 tile (25600B) global -> LDS.
    {
        const uint64_t gbase = (uint64_t)(uintptr_t)c20
                             + (uint64_t)blockIdx.x * (BLK * IN_W * 4);
        const uint32_t lbase = (uint32_t)(uintptr_t)tile;  // low 32b = LDS offset
        #pragma unroll
        for (int i = 0; i < 5; ++i) {
            uint32_t v  = (uint32_t)tid + (uint32_t)i * BLK;
            uint64_t ga = gbase + (uint64_t)v * 16u;
            uint32_t la = lbase + v * 16u;
            asm volatile("global_load_async_to_lds_b128 %0, %1, off"
                         :: "v"(la), "v"(ga) : "memory");
        }
        asm volatile("s_wait_asynccnt 0" ::: "memory");
    }
    __syncthreads();

    // -- Stage 2: shared layer-1 sums/diffs straight from the LDS row.
    float s1[10], d1[10];
    {
        const float4* rp = (const float4*)&tile[tid * IN_W];  // 80B row, 16B aligned
        #pragma unroll
        for (int i = 0; i < 5; ++i) {
            float4 q = rp[i];
            s1[2*i+0] = q.x + q.y;  d1[2*i+0] = q.x - q.y;
            s1[2*i+1] = q.z + q.w;  d1[2*i+1] = q.z - q.w;
        }
    }
    __syncthreads();   // all inputs consumed; tile now free for outputs

    // -- Stage 3: 19 trees, ~41 VALU each (~800 VALU/sample total).
    eval_forest(s1, d1, cw, &tile[tid * OUT_W]);
    __syncthreads();

    // -- Stage 4: async drain 320x19 f32 results (24320B) LDS -> global.
    {
        const uint64_t gbase = (uint64_t)(uintptr_t)out
                             + (uint64_t)blockIdx.x * (BLK * OUT_W * 4);
        const uint32_t lbase = (uint32_t)(uintptr_t)tile;
        #pragma unroll
        for (int i = 0; i < 4; ++i) {
            uint32_t v  = (uint32_t)tid + (uint32_t)i * BLK;
            uint64_t ga = gbase + (uint64_t)v * 16u;
            uint32_t la = lbase + v * 16u;
            asm volatile("global_store_async_from_lds_b128 %0, %1, off"
                         :: "v"(ga), "v"(la) : "memory");
        }
        if (tid < (BLK * OUT_W / 4 - 4 * BLK)) {   // 1520 - 1280 = 240
            uint32_t v  = (uint32_t)tid + 4u * BLK;
            uint64_t ga = gbase + (uint64_t)v * 16u;
            uint32_t la = lbase + v * 16u;
            asm volatile("global_store_async_from_lds_b128 %0, %1, off"
                         :: "v"(ga), "v"(la) : "memory");
        }
        asm volatile("s_wait_asynccnt 0" ::: "memory");
    }
}

// ---- Tail kernel (only if B % 320 != 0; unused for B = 1e6) ---------------
__global__ void tree_tail_kernel(const float* __restrict__ c20,
                                 const float* __restrict__ cw_in,
                                 float* __restrict__ out,
                                 int start, int B)
{
    const float* __restrict__ cw =
        (const float*)__builtin_assume_aligned(cw_in, 256);
    int s = start + blockIdx.x * blockDim.x + threadIdx.x;
    if (s >= B) return;
    float s1[10], d1[10];
    #pragma unroll
    for (int j = 0; j < 10; ++j) {
        float a = c20[(size_t)s * IN_W + 2*j];
        float b = c20[(size_t)s * IN_W + 2*j + 1];
        s1[j] = a + b;  d1[j] = a - b;
    }
    float r[OUT_W];
    eval_forest(s1, d1, cw, r);
    #pragma unroll
    for (int t = 0; t < OUT_W; ++t)
        out[(size_t)s * OUT_W + t] = r[t];
}

extern "C" void kernel_launch(void* const* d_in, const int* in_sizes, int n_in,
                              void* d_out, int out_size, void* d_ws, size_t ws_size,
                              hipStream_t stream)
{
    // setup_inputs order: p1 (int), p2 (int), c20 (f32, B*20), w_raw (f32, 361)
    const float* c20   = (const float*)d_in[2];
    const float* w_raw = (const float*)d_in[3];
    float* out = (float*)d_out;
    float* cw  = (float*)d_ws;          // 19*32 floats of scratch (2432B)

    const int B     = in_sizes[2] / IN_W;   // 1,000,000
    const int nfull = B / BLK;              // 3125
    const int tail  = B - nfull * BLK;      // 0

    sigmoid_prep_kernel<<<1, 384, 0, stream>>>(w_raw, cw);
    if (nfull > 0)
        tree_forward_kernel<<<nfull, BLK, 0, stream>>>(c20, cw, out);
    if (tail > 0)
        tree_tail_kernel<<<(tail + 255) / 256, 256, 0, stream>>>(
            c20, cw, out, nfull * BLK, B);
}